// FastrayTransformer_24756191494183
// MI455X (gfx1250) — compile-verified
//
#include <hip/hip_runtime.h>
#include <math.h>

typedef __attribute__((ext_vector_type(2))) float v2f;
typedef __attribute__((ext_vector_type(8))) float v8f;
typedef __attribute__((ext_vector_type(4))) int   v4i;

#define Dn     59
#define COUT   64
#define CTOT   123          // Dn + COUT
#define CPAD   128
#define XSTR   (CPAD + 1)   // padded LDS row stride (bank-conflict-free transpose)
#define Xd     128
#define Yd     128
#define Zd     7
#define Bdim   4
#define Ndim   6
#define CIN    256
#define Hc     16
#define Wc     44
#define HWc    (Hc * Wc)    // 704
#define BEV_ELEMS ((size_t)Bdim * COUT * Yd * Xd)

// ---- CDNA5 async global<->LDS path (ASYNCcnt-tracked), with fallback ------
#if defined(__has_builtin)
#if __has_builtin(__builtin_amdgcn_global_load_async_to_lds_b128) && \
    __has_builtin(__builtin_amdgcn_global_store_async_from_lds_b32)
#define HAVE_ASYNC_LDS 1
#endif
#endif
#ifndef HAVE_ASYNC_LDS
#define HAVE_ASYNC_LDS 0
#endif

typedef __attribute__((address_space(1))) v4i g_v4i;   // global int4
typedef __attribute__((address_space(3))) v4i l_v4i;   // LDS int4
typedef __attribute__((address_space(1))) int g_i32;   // global int
typedef __attribute__((address_space(3))) int l_i32;   // LDS int

__device__ __forceinline__ void wait_async0()
{
#if defined(__has_builtin) && __has_builtin(__builtin_amdgcn_s_wait_asynccnt)
    __builtin_amdgcn_s_wait_asynccnt(0);
#else
    asm volatile("s_wait_asynccnt 0" ::: "memory");
#endif
}

// ---------------------------------------------------------------------------
// Kernel 1: per-pixel projection GEMM (f32 WMMA 16x16x4) + bias + softmax.
// Block = (b,n, 16-pixel tile). 8 waves: wave w owns output channels [16w,16w+16).
// A-fragment = w_depth rows (K-contiguous per lane), B-fragment = LDS image tile.
// ---------------------------------------------------------------------------
__global__ __launch_bounds__(256)
void fastray_gemm_softmax(const float* __restrict__ img,
                          const float* __restrict__ wdep,
                          const float* __restrict__ bias,
                          float* __restrict__ depth_out,
                          float* __restrict__ feat_ws)
{
    __shared__ float lds_a[CIN * 16];    // [k][p] image tile, 16 KB
    __shared__ float lds_x[16 * XSTR];   // [p][o] result tile, ~8 KB
    __shared__ float lds_bias[CPAD];

    const int blk   = blockIdx.x;            // bn * 44 + mtile
    const int bn    = blk / (HWc / 16);
    const int pbase = (blk % (HWc / 16)) * 16;
    const int t     = threadIdx.x;
    const int lane  = t & 31;
    const int wv    = t >> 5;

    // Branchless bias -> LDS (padded with zeros up to 128).
    if (t < CPAD) {
        float bv = bias[min(t, CTOT - 1)];
        lds_bias[t] = (t < CTOT) ? bv : 0.0f;
    }

    // Stage the 16-pixel x 256-channel tile into LDS.
    const float* A = img + (size_t)bn * CIN * HWc + pbase;
#if HAVE_ASYNC_LDS
    {
        #pragma unroll
        for (int i = 0; i < 4; ++i) {
            int ch = t + i * 256;            // 16-byte chunk id, 1024 total
            int c  = ch >> 2;                // channel row (16 floats each)
            int p4 = (ch & 3) * 4;           // float offset within row
            __builtin_amdgcn_global_load_async_to_lds_b128(
                (g_v4i*)(A + (size_t)c * HWc + p4),
                (l_v4i*)(lds_a + ch * 4), 0, 0);
        }
        wait_async0();                       // this wave's async loads done
    }
#else
    #pragma unroll
    for (int i = 0; i < 16; ++i) {
        int idx = t + i * 256;               // idx = c*16 + p
        int c = idx >> 4, p = idx & 15;
        lds_a[idx] = A[(size_t)c * HWc + p];
    }
#endif
    __syncthreads();

    const int o0    = wv * 16;
    const int oc    = min(o0 + (lane & 15), CTOT - 1);  // clamp (no OOB read)
    const float* Wr = wdep + (size_t)oc * CIN;
    const int khalf = (lane >> 4) * 2;                  // K pair owned by this half-wave
    const int pcol  = lane & 15;

    v8f acc = {};
    for (int k = 0; k < CIN; k += 4) {
        v2f a, b;
        a.x = Wr[k + khalf];
        a.y = Wr[k + khalf + 1];
        b.x = lds_a[(k + khalf) * 16 + pcol];
        b.y = lds_a[(k + khalf + 1) * 16 + pcol];
        acc = __builtin_amdgcn_wmma_f32_16x16x4_f32(
                  false, a, false, b, (short)0, acc, false, false);
    }

    // Scatter D-tile (M=o, N=pixel) into LDS transpose buffer, adding bias.
    const int obase = o0 + 8 * (lane >> 4);
    #pragma unroll
    for (int r = 0; r < 8; ++r) {
        int o = obase + r;
        lds_x[pcol * XSTR + o] = acc[r] + lds_bias[o];
    }
    __syncthreads();

    // Softmax over bins [0,59): 16 threads per pixel, shfl-xor tree within 16 lanes.
    {
        const int pp = t >> 4;
        const int j  = t & 15;
        float xv[4];
        int   nb = 0;
        float m = -3.4e38f;
        for (int d = j; d < Dn; d += 16) {
            float x = lds_x[pp * XSTR + d];
            xv[nb++] = x;
            m = fmaxf(m, x);
        }
        #pragma unroll
        for (int s = 1; s < 16; s <<= 1) m = fmaxf(m, __shfl_xor(m, s, 16));
        float sum = 0.0f;
        for (int i = 0; i < nb; ++i) { xv[i] = __expf(xv[i] - m); sum += xv[i]; }
        #pragma unroll
        for (int s = 1; s < 16; s <<= 1) sum += __shfl_xor(sum, s, 16);
        const float inv = 1.0f / sum;
        const size_t dbase = ((size_t)bn * HWc + pbase + pp) * Dn;
        int k2 = 0;
        for (int d = j; d < Dn; d += 16) depth_out[dbase + d] = xv[k2++] * inv;
    }

    // Write the 64 feature channels to workspace (LDS -> global, no VGPR trip
    // on the async path).
#if HAVE_ASYNC_LDS
    #pragma unroll
    for (int i = 0; i < 4; ++i) {
        int idx = t + i * 256;               // idx = p*64 + o
        int pp = idx >> 6, o = idx & 63;
        float* g = feat_ws + ((size_t)bn * HWc + pbase + pp) * COUT + o;
        __builtin_amdgcn_global_store_async_from_lds_b32(
            (g_i32*)g, (l_i32*)(lds_x + pp * XSTR + Dn + o), 0, 0);
    }
    // implicit wait-idle at s_endpgm drains ASYNCcnt
#else
    #pragma unroll
    for (int i = 0; i < 4; ++i) {
        int idx = t + i * 256;               // idx = p*64 + o
        int pp = idx >> 6, o = idx & 63;
        feat_ws[((size_t)bn * HWc + pbase + pp) * COUT + o] =
            lds_x[pp * XSTR + Dn + o];
    }
#endif
}

// ---------------------------------------------------------------------------
// Kernel 2: voxel projection + gather + splat. Block = (b, y, 4 x's);
// thread = (x-lane, channel). Loops z (7) and camera n (6); gathers hit L2.
// ---------------------------------------------------------------------------
__global__ __launch_bounds__(256)
void fastray_splat(const float* __restrict__ depth,
                   const float* __restrict__ feat,
                   const float* __restrict__ c2e,
                   const float* __restrict__ Kin,
                   float* __restrict__ out)
{
    const int blk = blockIdx.x;                       // b * (Yd * Xd/4) + y * Xd/4 + xt
    const int b   = blk / (Yd * (Xd / 4));
    const int r   = blk % (Yd * (Xd / 4));
    const int y   = r / (Xd / 4);
    const int xt  = (r % (Xd / 4)) * 4;
    const int t   = threadIdx.x;
    const int c   = t & 63;
    const int x   = xt + (t >> 6);

    const float vx = -51.2f + 0.8f * (float)x;
    const float vy = -51.2f + 0.8f * (float)y;

    float acc = 0.0f;
    for (int n = 0; n < Ndim; ++n) {
        const float* M  = c2e + (size_t)(b * Ndim + n) * 16;  // row-major 4x4 cam2ego
        const float* Km = Kin + (size_t)(b * Ndim + n) * 9;   // row-major 3x3
        const float dx0 = vx - M[3];
        const float dy0 = vy - M[7];
        for (int z = 0; z < Zd; ++z) {
            const float vz  = -2.5f + 1.0f * (float)z;
            const float dz0 = vz - M[11];
            // cam = R^T (v - t)   (rigid inverse of cam2ego)
            const float cx = M[0] * dx0 + M[4] * dy0 + M[8]  * dz0;
            const float cy = M[1] * dx0 + M[5] * dy0 + M[9]  * dz0;
            const float cz = M[2] * dx0 + M[6] * dy0 + M[10] * dz0;
            const float zs = fmaxf(cz, 0.1f);
            const float nx = cx / zs, ny = cy / zs;
            const float fu = (Km[0] * nx + Km[1] * ny + Km[2]) * (1.0f / 16.0f);
            const float fv = (Km[3] * nx + Km[4] * ny + Km[5]) * (1.0f / 16.0f);
            const int   db = (int)(cz - 1.0f);                // trunc == astype(int32)
            const bool valid = (fu >= 0.0f) & (fu < (float)Wc) &
                               (fv >= 0.0f) & (fv < (float)Hc) &
                               (cz > 0.5f) & (db >= 0) & (db < Dn);
            const int u  = min(max((int)fu, 0), Wc - 1);
            const int v  = min(max((int)fv, 0), Hc - 1);
            const int d  = min(max(db, 0), Dn - 1);
            const size_t base = (size_t)(b * Ndim + n) * HWc + (size_t)(v * Wc + u);
            float wgt = depth[base * Dn + d];
            wgt = valid ? wgt : 0.0f;
            acc += wgt * feat[base * COUT + c];
        }
    }
    out[(((size_t)b * COUT + c) * Yd + y) * Xd + x] = acc;
}

// ---------------------------------------------------------------------------
extern "C" void kernel_launch(void* const* d_in, const int* in_sizes, int n_in,
                              void* d_out, int out_size, void* d_ws, size_t ws_size,
                              hipStream_t stream)
{
    const float* img  = (const float*)d_in[0];   // (B,N,256,16,44)
    const float* c2e  = (const float*)d_in[1];   // (B,N,4,4)
    const float* Kin  = (const float*)d_in[2];   // (B,N,3,3)
    const float* wdep = (const float*)d_in[3];   // (123,256)
    const float* bias = (const float*)d_in[4];   // (123,)

    float* bev_out   = (float*)d_out;                    // (B,64,128,128)
    float* depth_out = (float*)d_out + BEV_ELEMS;        // (B,N,16,44,59)
    float* feat_ws   = (float*)d_ws;                     // (B,N,704,64) scratch

    // Kernel 1: 24 (b,n) pairs * 44 pixel tiles
    fastray_gemm_softmax<<<dim3(Bdim * Ndim * (HWc / 16)), dim3(256), 0, stream>>>(
        img, wdep, bias, depth_out, feat_ws);

    // Kernel 2: B * Yd * (Xd/4) blocks
    fastray_splat<<<dim3(Bdim * Yd * (Xd / 4)), dim3(256), 0, stream>>>(
        depth_out, feat_ws, c2e, Kin, bev_out);
}